// VLAD_23098334118325
// MI455X (gfx1250) — compile-verified
//
#include <hip/hip_runtime.h>

typedef __attribute__((ext_vector_type(2))) float v2f;
typedef __attribute__((ext_vector_type(8))) float v8f;

namespace {
constexpr int B_ = 32;
constexpr int N_ = 8192;
constexpr int K_ = 128;
constexpr int D_ = 128;
constexpr int ITERS = 4;                         // desc-tiles per wave
constexpr int WAVES = 8;                         // waves per block (256 thr)
constexpr int DESC_PER_BLOCK = WAVES * 16 * ITERS;   // 512
constexpr int BLOCKS_PER_BATCH = N_ / DESC_PER_BLOCK; // 16
constexpr int POWER_ITERS = 24;
}

// ---------------------------------------------------------------- zero ws ---
__global__ void vlad_zero_kernel(float* p, int n) {
    int i = blockIdx.x * blockDim.x + threadIdx.x;
    if (i < n) p[i] = 0.0f;
}

// --------------------------------------------------- centroids / c2 / -2cT ---
// grid: K_ blocks x D_ threads
__global__ void vlad_prep_kernel(const float* __restrict__ csums,
                                 const float* __restrict__ pops,
                                 float* __restrict__ centroids,     // [K][D]
                                 float* __restrict__ centT_neg2,    // [D][K] = -2*c
                                 float* __restrict__ c2) {          // [K]
    int k = blockIdx.x;
    int d = threadIdx.x;
    float c = csums[k * D_ + d] / pops[k];
    centroids[k * D_ + d] = c;
    centT_neg2[d * K_ + k] = -2.0f * c;
    __shared__ float red[D_];
    red[d] = c * c;
    __syncthreads();
    for (int s = 64; s > 0; s >>= 1) {
        if (d < s) red[d] += red[d + s];
        __syncthreads();
    }
    if (d == 0) c2[k] = red[0];
}

// ------------------------------------------- WMMA distance + argmin + scatter
// grid: B_*BLOCKS_PER_BATCH blocks x 256 threads
__global__ void __launch_bounds__(256)
__attribute__((amdgpu_waves_per_eu(4)))
vlad_assign_kernel(const float* __restrict__ descs,        // [B][N][D]
                   const float* __restrict__ centT_neg2,   // [D][K]
                   const float* __restrict__ c2,           // [K]
                   float* __restrict__ desc_sums_g,        // [B][K][D]
                   float* __restrict__ pops_g) {           // [B][K]
    __shared__ float lds_sums[K_ * D_];   // exactly 64 KB
    for (int i = threadIdx.x; i < K_ * D_; i += blockDim.x) lds_sums[i] = 0.0f;
    __syncthreads();

    const int wave = threadIdx.x >> 5;
    const int lane = threadIdx.x & 31;
    const int laneHalf = lane >> 4;     // 0: lanes 0-15, 1: lanes 16-31
    const int q = lane & 15;

    const int b = blockIdx.x / BLOCKS_PER_BATCH;
    const int blk = blockIdx.x % BLOCKS_PER_BATCH;
    const int base_n = blk * DESC_PER_BLOCK;

#pragma unroll 1
    for (int it = 0; it < ITERS; ++it) {
        const int rowBase = base_n + (it * WAVES + wave) * 16;   // within batch

        // ---- load A fragments (16x4 f32 per step): lane holds row M=q,
        //      k-offsets laneHalf*2 + {0,1} within each 4-wide d-step.
        v2f a[32];
        const float* drow =
            descs + ((size_t)b * N_ + rowBase + q) * D_ + laneHalf * 2;
#pragma unroll
        for (int s = 0; s < 32; ++s) a[s] = *(const v2f*)(drow + s * 4);

        // ---- running argmin state per C-row (v = row % 8)
        float bestS[8];
        int bestK[8];
#pragma unroll
        for (int v = 0; v < 8; ++v) { bestS[v] = 3.4e38f; bestK[v] = 0; }

        // ---- k-tiles processed one at a time: single live accumulator,
        //      argmin folded in so nothing persists across tiles.
#pragma unroll 1
        for (int t = 0; t < 8; ++t) {
            float c2k = c2[t * 16 + q];
            v8f acc;                       // pre-load ||c||^2: acc = c2 - 2 x.c
#pragma unroll
            for (int e = 0; e < 8; ++e) acc[e] = c2k;

            const float* bcol = centT_neg2 + t * 16 + q;   // column k=t*16+q
#pragma unroll
            for (int s = 0; s < 32; ++s) {
                const int d0 = s * 4 + laneHalf * 2;
                v2f bb;
                bb.x = bcol[(d0 + 0) * K_];
                bb.y = bcol[(d0 + 1) * K_];
                acc = __builtin_amdgcn_wmma_f32_16x16x4_f32(
                    false, a[s], false, bb, (short)0, acc, false, false);
            }

            const int kk = t * 16 + q;
#pragma unroll
            for (int v = 0; v < 8; ++v) {
                float sv = acc[v];
                // t ascending + strict '<'  => lowest k kept on tie
                if (sv < bestS[v]) { bestS[v] = sv; bestK[v] = kk; }
            }
        }

        // ---- per-row argmin across lanes + scatter-add into LDS ----
#pragma unroll
        for (int v = 0; v < 8; ++v) {
            float bs = bestS[v];
            int bk = bestK[v];
            // butterfly min inside each 16-lane half (wave32)
#pragma unroll
            for (int off = 1; off < 16; off <<= 1) {
                float os = __shfl_xor(bs, off, 32);
                int ok = __shfl_xor(bk, off, 32);
                if (os < bs || (os == bs && ok < bk)) { bs = os; bk = ok; }
            }
            // row handled by this half-wave
            const int row = rowBase + v + laneHalf * 8;
            const float* dr = descs + ((size_t)b * N_ + row) * D_;
#pragma unroll
            for (int j = 0; j < 8; ++j) {
                int col = q + j * 16;
                atomicAdd(&lds_sums[bk * D_ + col], dr[col]);
            }
            if (q == 0) atomicAdd(&pops_g[b * K_ + bk], 1.0f);
        }
    }

    // ---- merge block-local sums into global ----
    __syncthreads();
    for (int i = threadIdx.x; i < K_ * D_; i += blockDim.x)
        atomicAdd(&desc_sums_g[(size_t)b * K_ * D_ + i], lds_sums[i]);
}

// ------------------------------------------------------------- residuals ---
// grid: B_*K_ blocks x D_ threads
__global__ void vlad_residual_kernel(const float* __restrict__ centroids,
                                     const float* __restrict__ desc_sums_g,
                                     const float* __restrict__ pops_g,
                                     float* __restrict__ out) {
    int bk = blockIdx.x;
    int d = threadIdx.x;
    int b = bk / K_;
    int k = bk % K_;
    float p = pops_g[b * K_ + k];
    out[(size_t)bk * D_ + d] =
        centroids[k * D_ + d] * p - desc_sums_g[(size_t)bk * D_ + d];
}

// -------------------------------- spectral norm (power iteration) + scale ---
// grid: B_ blocks x 128 threads; operates in place on d_out slice (L2-hot)
__global__ void vlad_specnorm_kernel(float* __restrict__ out) {
    __shared__ float vsh[D_];
    __shared__ float ush[K_];
    __shared__ float red[D_];
    const int b = blockIdx.x;
    const int t = threadIdx.x;
    float* M = out + (size_t)b * K_ * D_;

    float v = 1.0f + 0.001f * (float)t;   // generic init
    for (int it = 0; it < POWER_ITERS; ++it) {
        vsh[t] = v;
        __syncthreads();
        float s = 0.0f;                    // u = R v      (row dot)
#pragma unroll 4
        for (int j = 0; j < D_; ++j) s += M[t * D_ + j] * vsh[j];
        ush[t] = s;
        __syncthreads();
        float w = 0.0f;                    // w = R^T u    (col dot, coalesced)
#pragma unroll 4
        for (int i = 0; i < K_; ++i) w += M[i * D_ + t] * ush[i];
        red[t] = w * w;
        __syncthreads();
        for (int o = 64; o > 0; o >>= 1) {
            if (t < o) red[t] += red[t + o];
            __syncthreads();
        }
        float nrm = sqrtf(red[0]);
        __syncthreads();
        v = w / fmaxf(nrm, 1e-30f);
    }
    // sigma = ||R v||
    vsh[t] = v;
    __syncthreads();
    float s = 0.0f;
#pragma unroll 4
    for (int j = 0; j < D_; ++j) s += M[t * D_ + j] * vsh[j];
    __syncthreads();
    red[t] = s * s;
    __syncthreads();
    for (int o = 64; o > 0; o >>= 1) {
        if (t < o) red[t] += red[t + o];
        __syncthreads();
    }
    float inv = 1.0f / fmaxf(sqrtf(red[0]), 1e-30f);
    __syncthreads();
    for (int i = 0; i < K_; ++i) M[i * D_ + t] *= inv;
}

// ------------------------------------------------------------------ launch ---
extern "C" void kernel_launch(void* const* d_in, const int* in_sizes, int n_in,
                              void* d_out, int out_size, void* d_ws, size_t ws_size,
                              hipStream_t stream) {
    const float* descs = (const float*)d_in[0];   // [B][N][D]
    const float* csums = (const float*)d_in[1];   // [K][D]
    const float* pops  = (const float*)d_in[2];   // [K]
    float* out = (float*)d_out;                   // [B][K][D]

    float* ws = (float*)d_ws;
    float* centroids  = ws;                       // 16384
    float* centT_neg2 = ws + K_ * D_;             // 16384
    float* c2         = ws + 2 * K_ * D_;         // 128
    float* desc_sums  = c2 + K_;                  // 524288
    float* pops_g     = desc_sums + (size_t)B_ * K_ * D_;  // 4096

    // zero accumulators (desc_sums + pops are contiguous)
    int nz = B_ * K_ * D_ + B_ * K_;
    vlad_zero_kernel<<<(nz + 255) / 256, 256, 0, stream>>>(desc_sums, nz);

    vlad_prep_kernel<<<K_, D_, 0, stream>>>(csums, pops, centroids, centT_neg2, c2);

    vlad_assign_kernel<<<B_ * BLOCKS_PER_BATCH, WAVES * 32, 0, stream>>>(
        descs, centT_neg2, c2, desc_sums, pops_g);

    vlad_residual_kernel<<<B_ * K_, D_, 0, stream>>>(centroids, desc_sums, pops_g, out);

    vlad_specnorm_kernel<<<B_, D_, 0, stream>>>(out);
}